// Skipgram_70927089926296
// MI455X (gfx1250) — compile-verified
//
#include <hip/hip_runtime.h>
#include <hip/hip_bf16.h>

// ---------------------------------------------------------------------------
// Skip-gram negative-sampling loss on MI455X (gfx1250, wave32).
//
// Strategy: per wave, process groups of 16 samples. The 11 dot products per
// sample (1 positive + 10 negatives, D=128) are computed as diagonals of
// 16x16 WMMA tiles using V_WMMA_F32_16X16X4_F32 chained over K=128 (32 ops).
// All data stays f32 (exact); the workload is L2-resident gather-bound
// (tables ~102 MB < 192 MB L2), so WMMA compute waste is irrelevant.
// ---------------------------------------------------------------------------

#define NSAMP   65536
#define NDIMS   128
#define KNEG    10
#define NGROUPS (NSAMP / 16)   // 4096
#define NBLOCKS 128
#define NTHREADS 256

typedef __attribute__((ext_vector_type(2))) float v2f;
typedef __attribute__((ext_vector_type(8))) float v8f;

__device__ __forceinline__ float logsig(float x) {
    // stable log(sigmoid(x)) = min(x,0) - log1p(exp(-|x|))
    return fminf(x, 0.0f) - log1pf(expf(-fabsf(x)));
}

__device__ __forceinline__ float diag8(v8f c, int r) {
    // select c[r], r in [0,8) — compiles to v_cndmask chain (no scratch)
    float v = c[0];
    v = (r == 1) ? c[1] : v;
    v = (r == 2) ? c[2] : v;
    v = (r == 3) ? c[3] : v;
    v = (r == 4) ? c[4] : v;
    v = (r == 5) ? c[5] : v;
    v = (r == 6) ? c[6] : v;
    v = (r == 7) ? c[7] : v;
    return v;
}

// One WMMA dot-chain: rows of `arow` (per-lane context-vector row pointer)
// against cached in-vector fragments bf[]. Returns full 16x16 accumulator.
__device__ __forceinline__ v8f wmma_chain(const float* __restrict__ arow,
                                          const v2f* __restrict__ bf) {
    v8f cc = {};
#pragma unroll
    for (int c = 0; c < 32; ++c) {
        v2f af = *(const v2f*)(arow + 4 * c);
        cc = __builtin_amdgcn_wmma_f32_16x16x4_f32(
            /*neg_a=*/false, af, /*neg_b=*/false, bf[c],
            /*c_mod=*/(short)0, cc, /*reuse_a=*/false, /*reuse_b=*/false);
    }
    return cc;
}

__global__ void __launch_bounds__(NTHREADS)
skipgram_loss_kernel(const int* __restrict__ input_idx,
                     const int* __restrict__ output_idx,
                     const int* __restrict__ neg_idx,
                     const float* __restrict__ W_in,
                     const float* __restrict__ W_out,
                     float* __restrict__ partials) {
    const int lane = threadIdx.x & 31;
    const int wid  = threadIdx.x >> 5;
    const int wavesPerBlock = blockDim.x >> 5;
    const int globalWave = blockIdx.x * wavesPerBlock + wid;
    const int totalWaves = gridDim.x * wavesPerBlock;

    // f32 WMMA fragment addressing (ISA 7.12.2):
    //   A 16x4:  lane -> row M = lane&15, K pair = 2*(lane>>4) + {0,1}
    //   B 4x16:  lane -> col N = lane&15, K pair = 2*(lane>>4) + {0,1}
    //   C 16x16: lanes 0-15: N=lane, M=reg; lanes 16-31: N=lane-16, M=8+reg
    const int s  = lane & 15;
    const int hi = lane >> 4;
    // diagonal D[j][j]: lanes 0-7 hold j=lane at reg=lane,
    //                   lanes 24-31 hold j=lane-16 at reg=lane-24
    const float dmask = ((lane < 8) || (lane >= 24)) ? 1.0f : 0.0f;
    const int r = (lane < 8) ? lane : ((lane >= 24) ? (lane - 24) : 0);

    float acc = 0.0f;

    for (int g = globalWave; g < NGROUPS; g += totalWaves) {
        const int n = g * 16 + s;   // this lane's sample within the group

        // Cache B fragments (the 16 in-vectors) for all 32 K-chunks: 64 VGPRs.
        const float* brow = W_in + (size_t)input_idx[n] * NDIMS + 2 * hi;
        v2f bf[32];
#pragma unroll
        for (int c = 0; c < 32; ++c) bf[c] = *(const v2f*)(brow + 4 * c);

        // Positive pair: dot(in_n, out_n) -> log_sigmoid(d)
        {
            const float* arow = W_out + (size_t)output_idx[n] * NDIMS + 2 * hi;
            float d = diag8(wmma_chain(arow, bf), r);
            acc += dmask * logsig(d);
        }
        // Negatives: log_sigmoid(-dot(in_n, neg_{n,k}))
        for (int k = 0; k < KNEG; ++k) {
            const float* arow =
                W_out + (size_t)neg_idx[n * KNEG + k] * NDIMS + 2 * hi;
            float d = diag8(wmma_chain(arow, bf), r);
            acc += dmask * logsig(-d);
        }
    }

    // Wave reduction (wave32)
#pragma unroll
    for (int off = 16; off > 0; off >>= 1) acc += __shfl_xor(acc, off, 32);

    __shared__ float wsum[NTHREADS / 32];
    if (lane == 0) wsum[wid] = acc;
    __syncthreads();
    if (threadIdx.x == 0) {
        float b = 0.0f;
        for (int i = 0; i < wavesPerBlock; ++i) b += wsum[i];
        partials[blockIdx.x] = b;
    }
}

__global__ void reduce_partials_kernel(const float* __restrict__ partials,
                                       int nparts,
                                       float* __restrict__ out) {
    float a = 0.0f;
    for (int i = threadIdx.x; i < nparts; i += blockDim.x) a += partials[i];
#pragma unroll
    for (int off = 16; off > 0; off >>= 1) a += __shfl_xor(a, off, 32);

    __shared__ float wsum[8];
    const int lane = threadIdx.x & 31;
    const int wid  = threadIdx.x >> 5;
    if (lane == 0) wsum[wid] = a;
    __syncthreads();
    if (threadIdx.x == 0) {
        float s = 0.0f;
        for (int i = 0; i < (int)(blockDim.x >> 5); ++i) s += wsum[i];
        out[0] = s / (float)NSAMP;   // mean over samples
    }
}

extern "C" void kernel_launch(void* const* d_in, const int* in_sizes, int n_in,
                              void* d_out, int out_size, void* d_ws, size_t ws_size,
                              hipStream_t stream) {
    const int*   input_idx  = (const int*)d_in[0];
    const int*   output_idx = (const int*)d_in[1];
    const int*   neg_idx    = (const int*)d_in[2];
    const float* W_in       = (const float*)d_in[3];
    const float* W_out      = (const float*)d_in[4];
    float*       out        = (float*)d_out;
    float*       partials   = (float*)d_ws;   // NBLOCKS floats of scratch

    skipgram_loss_kernel<<<NBLOCKS, NTHREADS, 0, stream>>>(
        input_idx, output_idx, neg_idx, W_in, W_out, partials);
    reduce_partials_kernel<<<1, 256, 0, stream>>>(partials, NBLOCKS, out);
}